// Pool255_3410204033162
// MI455X (gfx1250) — compile-verified
//
#include <hip/hip_runtime.h>

// Pool255: sum of 8 directional cumulative-max pools, scaled by 1/8.
// One workgroup per 128x128 image; image staged to LDS via CDNA5 async
// global->LDS copies; 8 line scans accumulate via ds_add_f32 atomics.

#define HH 128
#define WW 128
#define LS 129              // padded LDS row stride (floats): 129 % 64 == 1 -> conflict-free
#define NT 256
#define NEGINF (-3.402823466e+38f)

typedef float v4f __attribute__((ext_vector_type(4)));

#define AS_GLOBAL __attribute__((address_space(1)))
#define AS_LDS    __attribute__((address_space(3)))

#if __has_builtin(__builtin_amdgcn_global_load_async_to_lds_b32)
#define HAVE_ASYNC_LDS 1
#else
#define HAVE_ASYNC_LDS 0
#endif

__launch_bounds__(NT)
__global__ void pool255_kernel(const float* __restrict__ in, float* __restrict__ out) {
  __shared__ __align__(16) float xs[HH * LS];   // 66048 B  input tile (padded rows)
  __shared__ __align__(16) float acc[HH * LS];  // 66048 B  pool-sum accumulator

  const int tid = threadIdx.x;
  const size_t img = blockIdx.x;
  const float* gin = in + img * (size_t)(HH * WW);
  float* gout = out + img * (size_t)(HH * WW);

  // ---- Stage tile into LDS (async, HBM -> LDS, no VGPR round trip) ----
#if HAVE_ASYNC_LDS
#pragma unroll 8
  for (int k = 0; k < 64; ++k) {
    const int e = tid + k * NT;           // element 0..16383 (lane-contiguous)
    const int r = e >> 7, c = e & 127;
    __builtin_amdgcn_global_load_async_to_lds_b32(
        (AS_GLOBAL int*)(gin + e),
        (AS_LDS int*)(xs + r * LS + c), 0, 0);
  }
#else
#pragma unroll 8
  for (int k = 0; k < 64; ++k) {
    const int e = tid + k * NT;
    xs[(e >> 7) * LS + (e & 127)] = gin[e];
  }
#endif

  // Zero the accumulator while the async copy is in flight.
  for (int i = tid; i < HH * LS; i += NT) acc[i] = 0.0f;

#if HAVE_ASYNC_LDS
#if __has_builtin(__builtin_amdgcn_s_wait_asynccnt)
  __builtin_amdgcn_s_wait_asynccnt(0);
#else
  asm volatile("s_wait_asynccnt 0" ::: "memory");
#endif
#endif
  __syncthreads();

  // ---- Rows: prefix max (dir (0,-1)) + suffix max (dir (0,1)) ----
  // Waves 0-3 forward on row tid, waves 4-7 backward on row tid-128.
  if (tid < HH) {
    const int base = tid * LS;
    float m = NEGINF;
#pragma unroll 4
    for (int j = 0; j < WW; ++j) {
      m = fmaxf(m, xs[base + j]);
      atomicAdd(&acc[base + j], m);     // ds_add_f32, no return
    }
  } else {
    const int base = (tid - HH) * LS;
    float m = NEGINF;
#pragma unroll 4
    for (int j = WW - 1; j >= 0; --j) {
      m = fmaxf(m, xs[base + j]);
      atomicAdd(&acc[base + j], m);
    }
  }

  // ---- Columns: prefix max (dir (-1,0)) + suffix max (dir (1,0)) ----
  if (tid < WW) {
    const int c = tid;
    float m = NEGINF;
#pragma unroll 4
    for (int i = 0; i < HH; ++i) {
      m = fmaxf(m, xs[i * LS + c]);
      atomicAdd(&acc[i * LS + c], m);
    }
  } else {
    const int c = tid - WW;
    float m = NEGINF;
#pragma unroll 4
    for (int i = HH - 1; i >= 0; --i) {
      m = fmaxf(m, xs[i * LS + c]);
      atomicAdd(&acc[i * LS + c], m);
    }
  }

  // ---- Main diagonals (i-j = const): dirs (-1,-1) prefix and (1,1) suffix ----
  // Thread t scans line t forward and line (t+127)%255 backward; the pairing
  // keeps per-thread work ~129 elements (length-balanced).
  if (tid < 255) {
    {
      const int off = tid - 127;
      const int i0 = off >= 0 ? off : 0;
      const int j0 = off >= 0 ? 0 : -off;
      const int len = HH - (off >= 0 ? off : -off);
      int idx = i0 * LS + j0;
      float m = NEGINF;
      for (int k = 0; k < len; ++k) {
        m = fmaxf(m, xs[idx]);
        atomicAdd(&acc[idx], m);
        idx += LS + 1;
      }
    }
    {
      const int t2 = (tid + 127) % 255;
      const int off = t2 - 127;
      const int i0 = off >= 0 ? off : 0;
      const int j0 = off >= 0 ? 0 : -off;
      const int len = HH - (off >= 0 ? off : -off);
      int idx = (i0 + len - 1) * LS + (j0 + len - 1);
      float m = NEGINF;
      for (int k = 0; k < len; ++k) {
        m = fmaxf(m, xs[idx]);
        atomicAdd(&acc[idx], m);
        idx -= LS + 1;
      }
    }
  }

  // ---- Anti-diagonals (i+j = const): dirs (-1,1) prefix and (1,-1) suffix ----
  if (tid < 255) {
    {
      const int s = tid;
      const int i0 = s > 127 ? s - 127 : 0;
      const int j0 = s - i0;
      const int len = HH - (s >= 127 ? s - 127 : 127 - s);
      int idx = i0 * LS + j0;
      float m = NEGINF;
      for (int k = 0; k < len; ++k) {
        m = fmaxf(m, xs[idx]);
        atomicAdd(&acc[idx], m);
        idx += LS - 1;
      }
    }
    {
      const int s = (tid + 127) % 255;
      const int i0 = s > 127 ? s - 127 : 0;
      const int j0 = s - i0;
      const int len = HH - (s >= 127 ? s - 127 : 127 - s);
      int idx = (i0 + len - 1) * LS + (j0 - len + 1);
      float m = NEGINF;
      for (int k = 0; k < len; ++k) {
        m = fmaxf(m, xs[idx]);
        atomicAdd(&acc[idx], m);
        idx -= LS - 1;
      }
    }
  }

  __syncthreads();

  // ---- Scale by 1/8 and stream out (write-once: non-temporal b128 stores) ----
#pragma unroll 4
  for (int k = 0; k < 16; ++k) {
    const int c4 = tid + k * NT;          // float4 chunk 0..4095
    const int r = c4 >> 5;                // 32 chunks per row
    const int cc = (c4 & 31) << 2;
    const int b = r * LS + cc;
    v4f v = {acc[b], acc[b + 1], acc[b + 2], acc[b + 3]};
    v *= 0.125f;
    __builtin_nontemporal_store(v, (v4f*)(gout + ((size_t)c4 << 2)));
  }
}

extern "C" void kernel_launch(void* const* d_in, const int* in_sizes, int n_in,
                              void* d_out, int out_size, void* d_ws, size_t ws_size,
                              hipStream_t stream) {
  (void)n_in; (void)out_size; (void)d_ws; (void)ws_size;
  const float* x = (const float*)d_in[0];
  float* out = (float*)d_out;
  const int nimg = in_sizes[0] / (HH * WW);   // 8*256 = 2048 images
  pool255_kernel<<<dim3(nimg), dim3(NT), 0, stream>>>(x, out);
}